// LocalAttention_47313359733390
// MI455X (gfx1250) — compile-verified
//
#include <hip/hip_runtime.h>
#include <hip/hip_bf16.h>

typedef __attribute__((ext_vector_type(2))) float v2f;
typedef __attribute__((ext_vector_type(8))) float v8f;

#define WSZ   128            // window size
#define EDIM  64             // head dim
#define KROWS 256            // extended key rows (look-back + current)
#define LDK   68             // padded LDS row stride (floats) for K/V
#define PSTR  18             // padded stride for per-wave P staging tile (even -> 8B aligned)
#define NWAVE 8

__global__ __launch_bounds__(256)
void LocalAttention_47313359733390_kernel(const float* __restrict__ q,
                                          const float* __restrict__ k,
                                          const float* __restrict__ v,
                                          float* __restrict__ out) {
    extern __shared__ float smem[];
    float* K_lds = smem;                       // 256*68
    float* V_lds = K_lds + KROWS * LDK;        // 256*68
    float* P_lds = V_lds + KROWS * LDK;        // 8*16*18

    const int blk  = blockIdx.x;               // 0..2047
    const int w    = blk & 63;                 // window index
    const int Bh   = blk >> 6;                 // merged batch*head
    const int tid  = threadIdx.x;
    const int wave = tid >> 5;
    const int lane = tid & 31;
    const int m    = lane & 15;                // row/col-in-tile index
    const int kh   = lane >> 4;                // K-half select (lanes 0-15 vs 16-31)

    const size_t base = ((size_t)Bh * 64 + (size_t)w) * (WSZ * EDIM);

    // ---- async global->LDS stage of K_ext / V_ext (ASYNCcnt path) ----
    for (int i = tid; i < KROWS * (EDIM / 4); i += 256) {
        const int row = i >> 4;                // 0..255
        const int c4  = i & 15;                // float4 column
        const unsigned loff = (unsigned)((row * LDK + c4 * 4) * sizeof(float));
        if (w == 0 && row < WSZ) {
            const float4 z = make_float4(0.f, 0.f, 0.f, 0.f);
            *(float4*)(&K_lds[row * LDK + c4 * 4]) = z;
            *(float4*)(&V_lds[row * LDK + c4 * 4]) = z;
        } else {
            const long off = (long)row - WSZ;  // relative to current window start
            const float* gk = k + base + off * EDIM + c4 * 4;
            const float* gv = v + base + off * EDIM + c4 * 4;
            asm volatile("global_load_async_to_lds_b128 %0, %1, off"
                         :: "v"(loff), "v"(gk) : "memory");
            asm volatile("global_load_async_to_lds_b128 %0, %1, off"
                         :: "v"((unsigned)(loff + KROWS * LDK * sizeof(float))),
                            "v"(gv) : "memory");
        }
    }

    // ---- load Q A-fragments (scale folded in; 0.125 = 2^-3 is exact) ----
    const float* qrow = q + base + (size_t)(wave * 16 + m) * EDIM;
    v2f qa[16];
#pragma unroll
    for (int c = 0; c < 16; ++c) {
        qa[c].x = qrow[4 * c + 2 * kh] * 0.125f;
        qa[c].y = qrow[4 * c + 2 * kh + 1] * 0.125f;
    }

    asm volatile("s_wait_asynccnt 0x0" ::: "memory");
    __syncthreads();

    // ---- S = (Q*scale) * K_ext^T : batch 16 B-frag loads, then dense WMMA chain ----
    v8f S[16];
#pragma unroll
    for (int nt = 0; nt < 16; ++nt) {
        const float* kb = &K_lds[(nt * 16 + m) * LDK + 2 * kh];
        v2f bf[16];
#pragma unroll
        for (int c = 0; c < 16; ++c)
            bf[c] = *(const v2f*)(kb + 4 * c);      // ds_load_b64, 8B aligned
        v8f acc = {};
#pragma unroll
        for (int c = 0; c < 16; ++c)
            acc = __builtin_amdgcn_wmma_f32_16x16x4_f32(
                false, qa[c], false, bf[c], (short)0, acc, false, false);
        S[nt] = acc;
    }

    // ---- mask + numerically-stable softmax (rows live in 16-lane halves) ----
    const float NEG = -3.402823466e38f;
#pragma unroll
    for (int vg = 0; vg < 8; ++vg) {
        const int i = wave * 16 + vg + 8 * kh;      // q row within window (0..127)
        float lm = NEG;
#pragma unroll
        for (int nt = 0; nt < 16; ++nt) {
            const int j = nt * 16 + m;              // extended key column (0..255)
            float s = S[nt][vg];
            const bool masked = (w == 0 && j < 128) || (j >= 128 && i < (j - 128));
            s = masked ? NEG : s;
            S[nt][vg] = s;
            lm = fmaxf(lm, s);
        }
        for (int off = 1; off < 16; off <<= 1)
            lm = fmaxf(lm, __shfl_xor(lm, off, 32));
        float ls = 0.f;
#pragma unroll
        for (int nt = 0; nt < 16; ++nt) {
            const float p = __expf(S[nt][vg] - lm);
            S[nt][vg] = p;
            ls += p;
        }
        for (int off = 1; off < 16; off <<= 1)
            ls += __shfl_xor(ls, off, 32);
        const float inv = 1.0f / ls;
#pragma unroll
        for (int nt = 0; nt < 16; ++nt) S[nt][vg] *= inv;
    }

    // ---- O = P * V_ext : per key-tile, C/D->A layout flip via per-wave LDS,
    //      loads batched ahead of a dense 16-WMMA chain ----
    v8f O[4] = {};
    float* pw = &P_lds[wave * 16 * PSTR];
#pragma unroll
    for (int kt = 0; kt < 16; ++kt) {
#pragma unroll
        for (int vg = 0; vg < 8; ++vg)
            pw[(vg + 8 * kh) * PSTR + m] = S[kt][vg];
        // per-wave buffer; LDS ops from one wave are in-order -> no barrier
        v2f a[4], bb[16];
#pragma unroll
        for (int c = 0; c < 4; ++c)
            a[c] = *(const v2f*)(pw + m * PSTR + 4 * c + 2 * kh);
#pragma unroll
        for (int c = 0; c < 4; ++c)
#pragma unroll
            for (int et = 0; et < 4; ++et) {
                const float* vb = &V_lds[(kt * 16 + 4 * c + 2 * kh) * LDK + et * 16 + m];
                bb[c * 4 + et].x = vb[0];
                bb[c * 4 + et].y = vb[LDK];
            }
#pragma unroll
        for (int c = 0; c < 4; ++c)
#pragma unroll
            for (int et = 0; et < 4; ++et)
                O[et] = __builtin_amdgcn_wmma_f32_16x16x4_f32(
                    false, a[c], false, bb[c * 4 + et], (short)0, O[et], false, false);
    }

    // ---- stage O in (now dead) K_lds, then fully coalesced float4 stores ----
    __syncthreads();   // all waves done reading K_lds
#pragma unroll
    for (int et = 0; et < 4; ++et)
#pragma unroll
        for (int vg = 0; vg < 8; ++vg)
            K_lds[(wave * 16 + vg + 8 * kh) * LDK + et * 16 + m] = O[et][vg];
    __syncthreads();   // staged rows visible to all threads
    for (int i = tid; i < WSZ * (EDIM / 4); i += 256) {
        const int row = i >> 4;
        const int c4  = i & 15;
        const float4 val = *(const float4*)(&K_lds[row * LDK + c4 * 4]);
        *(float4*)(out + base + (size_t)row * EDIM + c4 * 4) = val;
    }
}

extern "C" void kernel_launch(void* const* d_in, const int* in_sizes, int n_in,
                              void* d_out, int out_size, void* d_ws, size_t ws_size,
                              hipStream_t stream) {
    const float* q = (const float*)d_in[0];
    const float* k = (const float*)d_in[1];
    const float* v = (const float*)d_in[2];
    float* out = (float*)d_out;

    const int blocks = 32 * 64;  // (batch*heads) * windows
    const size_t shmem = (size_t)(2 * KROWS * LDK + NWAVE * 16 * PSTR) * sizeof(float);
    LocalAttention_47313359733390_kernel<<<blocks, 256, shmem, stream>>>(q, k, v, out);
}